// Net_26414048870710
// MI455X (gfx1250) — compile-verified
//
#include <hip/hip_runtime.h>
#include <math.h>

#define NN 100000
#define EE 1000000
#define NODE_DIM 10
#define TIME_DIM 10
#define EDGE_DIM 16
#define EMB 20
#define EMBP 32     // padded column count in LDS (zeros in cols 20..31)
#define FULL_K 56   // EDGE_DIM + 2*(NODE_DIM+TIME_DIM)

typedef __attribute__((ext_vector_type(2))) float v2f;
typedef __attribute__((ext_vector_type(8))) float v8f;

#if defined(__has_builtin)
#if __has_builtin(__builtin_amdgcn_global_load_async_to_lds_b32) && \
    __has_builtin(__builtin_amdgcn_s_wait_asynccnt)
#define HAVE_ASYNC_LDS 1
#endif
#endif
#ifndef HAVE_ASYNC_LDS
#define HAVE_ASYNC_LDS 0
#endif

typedef __attribute__((address_space(1))) int* gas_i32p;
typedef __attribute__((address_space(3))) int* las_i32p;

__device__ __forceinline__ void lds_copy_f32(float* dst_lds, const float* src_glb) {
#if HAVE_ASYNC_LDS
  __builtin_amdgcn_global_load_async_to_lds_b32(
      (gas_i32p)src_glb, (las_i32p)dst_lds, 0, 0);
#else
  *dst_lds = *src_glb;
#endif
}
__device__ __forceinline__ void lds_copy_wait() {
#if HAVE_ASYNC_LDS
  __builtin_amdgcn_s_wait_asynccnt(0);
#endif
}

__device__ __forceinline__ void atomic_max_float(float* addr, float val) {
  // ordered-int trick; amax initialized to -inf
  if (val >= 0.0f) atomicMax((int*)addr, __float_as_int(val));
  else             atomicMin((unsigned int*)addr, __float_as_uint(val));
}

// ---------------- init / zero ----------------
__global__ void k_init(float* __restrict__ amax, float* __restrict__ denom) {
  int i = blockIdx.x * blockDim.x + threadIdx.x;
  if (i < NN) { amax[i] = -INFINITY; denom[i] = 0.0f; }
}

__global__ void k_zero(float* __restrict__ p, int n) {
  int i = blockIdx.x * blockDim.x + threadIdx.x;
  if (i < n) p[i] = 0.0f;
}

// ---------------- k-hop gather -> scatter-add ----------------
__global__ void k_khop(const float* __restrict__ in, float* __restrict__ out,
                       const int* __restrict__ ei, const int* __restrict__ hei) {
  int i = blockIdx.x * blockDim.x + threadIdx.x;
  if (i >= 2 * EE) return;
  int s, d;
  if (i < EE) { s = ei[i];       d = ei[EE + i]; }
  else        { s = hei[i - EE]; d = hei[EE + (i - EE)]; }
  const float* xs = in + (size_t)s * NODE_DIM;
  float* xd = out + (size_t)d * NODE_DIM;
#pragma unroll
  for (int j = 0; j < NODE_DIM; ++j) atomicAdd(&xd[j], xs[j]);
}

// ---------------- node linears q,k,v,skip via fp32 WMMA 16x16x4 ----------------
// Weights staged in LDS, padded to 32 cols with zeros -> branch-free B loads.
__global__ void k_node_linear(const float* __restrict__ z,
                              const float* __restrict__ Wq, const float* __restrict__ bq,
                              const float* __restrict__ Wk, const float* __restrict__ bk,
                              const float* __restrict__ Wv, const float* __restrict__ bv,
                              const float* __restrict__ Wskip, const float* __restrict__ bskip,
                              float* __restrict__ q, float* __restrict__ kk,
                              float* __restrict__ v, float* __restrict__ outskip) {
  __shared__ float sW[4 * EMB * EMBP];   // [m][k][col], zero-padded cols
  __shared__ float sB[4 * EMBP];         // [m][col], zero-padded

  const float* Ws[4]   = {Wq, Wk, Wv, Wskip};
  const float* bs[4]   = {bq, bk, bv, bskip};
  float*       outs[4] = {q, kk, v, outskip};

  // cooperative staging (async copy for valid cols, zero the pad)
  const int NW = 4 * EMB * EMBP;         // 2560
  for (int i = threadIdx.x; i < NW + 4 * EMBP; i += blockDim.x) {
    if (i < NW) {
      int m = i / (EMB * EMBP), rem = i % (EMB * EMBP);
      int k = rem / EMBP, c = rem % EMBP;
      if (c < EMB) lds_copy_f32(&sW[i], Ws[m] + k * EMB + c);
      else         sW[i] = 0.0f;
    } else {
      int j = i - NW;
      int m = j / EMBP, c = j % EMBP;
      if (c < EMB) lds_copy_f32(&sB[j], bs[m] + c);
      else         sB[j] = 0.0f;
    }
  }
  lds_copy_wait();
  __syncthreads();

  const int lane  = threadIdx.x & 31;
  const int tile  = blockIdx.x * (blockDim.x >> 5) + (threadIdx.x >> 5);
  if (tile >= NN / 16) return;           // wave-uniform -> EXEC all-ones for WMMA
  const int base  = tile * 16;
  const int khalf = lane >> 4;           // 0: K={k0,k0+1}; 1: K={k0+2,k0+3}

  // A fragments: 16 nodes x 20 (5 chunks of K=4), shared by all 4 weight matrices
  v2f A[5];
  const float* zr = z + (size_t)(base + (lane & 15)) * EMB;
#pragma unroll
  for (int c = 0; c < 5; ++c) {
    int k0 = 4 * c + 2 * khalf;
    A[c].x = zr[k0];
    A[c].y = zr[k0 + 1];
  }

  for (int m = 0; m < 4; ++m) {
    const float* Wm = &sW[m * EMB * EMBP];
#pragma unroll
    for (int nt = 0; nt < 2; ++nt) {     // N = 20 -> tiles of 16 + 4
      int col = nt * 16 + (lane & 15);
      float bias = sB[m * EMBP + col];   // 0 in pad region
      v8f acc = {bias, bias, bias, bias, bias, bias, bias, bias};
#pragma unroll
      for (int c = 0; c < 5; ++c) {
        int k0 = 4 * c + 2 * khalf;
        v2f B;
        B.x = Wm[(k0 + 0) * EMBP + col];
        B.y = Wm[(k0 + 1) * EMBP + col];
        acc = __builtin_amdgcn_wmma_f32_16x16x4_f32(false, A[c], false, B,
                                                    (short)0, acc, false, false);
      }
      if (col < EMB) {
        int mbase = base + ((lane >= 16) ? 8 : 0);
        float* o = outs[m];
#pragma unroll
        for (int r = 0; r < 8; ++r)
          o[(size_t)(mbase + r) * EMB + col] = acc[r];
      }
    }
  }
}

// ---------------- edge features (built in registers) @ We via fp32 WMMA ----------------
__device__ __forceinline__ float ea_elem(int k, const float* __restrict__ eattr,
                                         float tv,
                                         const float* __restrict__ ts,
                                         const float* __restrict__ td,
                                         const float* __restrict__ xs,
                                         const float* __restrict__ xd) {
  if (k < EDGE_DIM) return eattr[k];
  k -= EDGE_DIM;
  if (k < TIME_DIM) return tv - ts[k];
  k -= TIME_DIM;
  if (k < TIME_DIM) return tv - td[k];
  k -= TIME_DIM;
  if (k < NODE_DIM) return xs[k];
  return xd[k - NODE_DIM];
}

__global__ void k_edge_e(const float* __restrict__ edge_attr,
                         const float* __restrict__ t,
                         const float* __restrict__ encT,
                         const float* __restrict__ xh,
                         const int* __restrict__ src,
                         const int* __restrict__ dst,
                         const float* __restrict__ We,
                         float* __restrict__ e) {
  __shared__ float sWe[FULL_K * EMBP];   // 56 x 32, zero-padded cols

  for (int i = threadIdx.x; i < FULL_K * EMBP; i += blockDim.x) {
    int k = i / EMBP, c = i % EMBP;
    if (c < EMB) lds_copy_f32(&sWe[i], We + k * EMB + c);
    else         sWe[i] = 0.0f;
  }
  lds_copy_wait();
  __syncthreads();

  const int lane  = threadIdx.x & 31;
  const int tile  = blockIdx.x * (blockDim.x >> 5) + (threadIdx.x >> 5);
  if (tile >= EE / 16) return;
  const int base  = tile * 16;
  const int edge  = base + (lane & 15);
  const int khalf = lane >> 4;

  const int s = src[edge], d = dst[edge];
  const float tv = t[edge];
  const float* eattr = edge_attr + (size_t)edge * EDGE_DIM;
  const float* ts = encT + (size_t)s * TIME_DIM;
  const float* td = encT + (size_t)d * TIME_DIM;
  const float* xs = xh   + (size_t)s * NODE_DIM;
  const float* xd = xh   + (size_t)d * NODE_DIM;

  // A = ea tile, 16 edges x 56 (14 chunks of K=4), never materialized in memory
  v2f A[14];
#pragma unroll
  for (int c = 0; c < 14; ++c) {
    int k0 = 4 * c + 2 * khalf;
    A[c].x = ea_elem(k0,     eattr, tv, ts, td, xs, xd);
    A[c].y = ea_elem(k0 + 1, eattr, tv, ts, td, xs, xd);
  }

#pragma unroll
  for (int nt = 0; nt < 2; ++nt) {
    int col = nt * 16 + (lane & 15);
    v8f acc = {};                         // lin_edge has no bias
#pragma unroll
    for (int c = 0; c < 14; ++c) {
      int k0 = 4 * c + 2 * khalf;
      v2f B;
      B.x = sWe[(k0 + 0) * EMBP + col];
      B.y = sWe[(k0 + 1) * EMBP + col];
      acc = __builtin_amdgcn_wmma_f32_16x16x4_f32(false, A[c], false, B,
                                                  (short)0, acc, false, false);
    }
    if (col < EMB) {
      int mbase = base + ((lane >= 16) ? 8 : 0);
#pragma unroll
      for (int r = 0; r < 8; ++r)
        e[(size_t)(mbase + r) * EMB + col] = acc[r];
    }
  }
}

// ---------------- attention epilogue ----------------
__global__ void k_alpha(const float* __restrict__ q, const float* __restrict__ kk,
                        const float* __restrict__ e,
                        const int* __restrict__ src, const int* __restrict__ dst,
                        float* __restrict__ alpha, float* __restrict__ amax) {
  int i = blockIdx.x * blockDim.x + threadIdx.x;
  if (i >= EE) return;
  int s = src[i], d = dst[i];
  const float* qd = q  + (size_t)d * EMB;
  const float* ks = kk + (size_t)s * EMB;
  const float* ei = e  + (size_t)i * EMB;
  float a = 0.0f;
#pragma unroll
  for (int j = 0; j < EMB; ++j) a += qd[j] * (ks[j] + ei[j]);
  a *= 0.22360679774997896f;   // 1/sqrt(EMB)
  alpha[i] = a;
  atomic_max_float(&amax[d], a);
}

__global__ void k_expdenom(const int* __restrict__ dst,
                           const float* __restrict__ amax,
                           float* __restrict__ alpha, float* __restrict__ denom) {
  int i = blockIdx.x * blockDim.x + threadIdx.x;
  if (i >= EE) return;
  int d = dst[i];
  float ex = __expf(alpha[i] - amax[d]);
  alpha[i] = ex;                 // reuse buffer (rewritten every call)
  atomicAdd(&denom[d], ex);
}

__global__ void k_scatter_out(const int* __restrict__ src, const int* __restrict__ dst,
                              const float* __restrict__ v, const float* __restrict__ e,
                              const float* __restrict__ ex, const float* __restrict__ denom,
                              float* __restrict__ out) {
  int i = blockIdx.x * blockDim.x + threadIdx.x;
  if (i >= EE) return;
  int s = src[i], d = dst[i];
  float attn = ex[i] / (denom[d] + 1e-16f);
  const float* vs = v + (size_t)s * EMB;
  const float* ei = e + (size_t)i * EMB;
  float* od = out + (size_t)d * EMB;
#pragma unroll
  for (int j = 0; j < EMB; ++j) atomicAdd(&od[j], attn * (vs[j] + ei[j]));
}

// ---------------- launcher ----------------
extern "C" void kernel_launch(void* const* d_in, const int* in_sizes, int n_in,
                              void* d_out, int out_size, void* d_ws, size_t ws_size,
                              hipStream_t stream) {
  const float* x     = (const float*)d_in[0];
  const int*   src   = (const int*)  d_in[1];
  const int*   dst   = (const int*)  d_in[2];
  const int*   ei    = (const int*)  d_in[3];
  const int*   hei   = (const int*)  d_in[4];
  const float* eattr = (const float*)d_in[5];
  const float* t     = (const float*)d_in[6];
  const float* encT  = (const float*)d_in[7];
  const float* z     = (const float*)d_in[8];
  const float* Wq    = (const float*)d_in[9];  const float* bq    = (const float*)d_in[10];
  const float* Wk    = (const float*)d_in[11]; const float* bk    = (const float*)d_in[12];
  const float* Wv    = (const float*)d_in[13]; const float* bv    = (const float*)d_in[14];
  const float* We    = (const float*)d_in[15];
  const float* Wskip = (const float*)d_in[16]; const float* bskip = (const float*)d_in[17];
  float* out = (float*)d_out;

  float* ws = (float*)d_ws;
  size_t off = 0;
  float* xhA   = ws + off; off += (size_t)NN * NODE_DIM;
  float* xhB   = ws + off; off += (size_t)NN * NODE_DIM;
  float* q     = ws + off; off += (size_t)NN * EMB;
  float* kk    = ws + off; off += (size_t)NN * EMB;
  float* v     = ws + off; off += (size_t)NN * EMB;
  float* e     = ws + off; off += (size_t)EE * EMB;
  float* alpha = ws + off; off += (size_t)EE;
  float* amax  = ws + off; off += (size_t)NN;
  float* denom = ws + off; off += (size_t)NN;
  (void)off; (void)ws_size; (void)in_sizes; (void)n_in; (void)out_size;

  const int T = 256;

  // 3-hop aggregation over batch + historical edges (ping-pong xhA/xhB)
  k_zero<<<(NN * NODE_DIM + T - 1) / T, T, 0, stream>>>(xhA, NN * NODE_DIM);
  k_khop<<<(2 * EE + T - 1) / T, T, 0, stream>>>(x,   xhA, ei, hei);
  k_zero<<<(NN * NODE_DIM + T - 1) / T, T, 0, stream>>>(xhB, NN * NODE_DIM);
  k_khop<<<(2 * EE + T - 1) / T, T, 0, stream>>>(xhA, xhB, ei, hei);
  k_zero<<<(NN * NODE_DIM + T - 1) / T, T, 0, stream>>>(xhA, NN * NODE_DIM);
  k_khop<<<(2 * EE + T - 1) / T, T, 0, stream>>>(xhB, xhA, ei, hei);
  // xhA now holds xh

  k_init<<<(NN + T - 1) / T, T, 0, stream>>>(amax, denom);

  {
    int tiles  = NN / 16;                 // 6250 (exact)
    int wpb    = T / 32;                  // 8 waves / block
    int blocks = (tiles + wpb - 1) / wpb;
    k_node_linear<<<blocks, T, 0, stream>>>(z, Wq, bq, Wk, bk, Wv, bv, Wskip, bskip,
                                            q, kk, v, out);   // out <- z@Wskip + bskip
  }
  {
    int tiles  = EE / 16;                 // 62500 (exact)
    int wpb    = T / 32;
    int blocks = (tiles + wpb - 1) / wpb;
    k_edge_e<<<blocks, T, 0, stream>>>(eattr, t, encT, xhA, src, dst, We, e);
  }

  k_alpha<<<(EE + T - 1) / T, T, 0, stream>>>(q, kk, e, src, dst, alpha, amax);
  k_expdenom<<<(EE + T - 1) / T, T, 0, stream>>>(dst, amax, alpha, denom);
  k_scatter_out<<<(EE + T - 1) / T, T, 0, stream>>>(src, dst, v, e, alpha, denom, out);
}